// GraphSAGE_residual_25460566130852
// MI455X (gfx1250) — compile-verified
//
#include <hip/hip_runtime.h>
#include <hip/hip_bf16.h>

#define N_NODES 100000
#define N_EDGES 1600000
#define DIM     128
#define RW      0.001f

typedef float v2f __attribute__((ext_vector_type(2)));
typedef float v8f __attribute__((ext_vector_type(8)));

// ---------------------------------------------------------------------------
// Kernel 1: fold the affine blend into two weight matrices + one bias.
//   W1 = rw*Wl ;  W2 = rw*Wr + (1-rw)*Wres ;  bc = rw*bl + (1-rw)*bres
// ---------------------------------------------------------------------------
__global__ void combine_weights_kernel(const float* __restrict__ Wl,
                                       const float* __restrict__ bl,
                                       const float* __restrict__ Wr,
                                       const float* __restrict__ Wres,
                                       const float* __restrict__ bres,
                                       float* __restrict__ W1,
                                       float* __restrict__ W2,
                                       float* __restrict__ bc) {
    int i = blockIdx.x * blockDim.x + threadIdx.x;
    const float rw = RW, orw = 1.0f - RW;
    if (i < DIM * DIM) {
        W1[i] = rw * Wl[i];
        W2[i] = rw * Wr[i] + orw * Wres[i];
    }
    if (i < DIM) {
        bc[i] = rw * bl[i] + orw * bres[i];
    }
}

// ---------------------------------------------------------------------------
// Kernel 2: zero the aggregation buffer (float4 stores).
// ---------------------------------------------------------------------------
__global__ void zero_kernel(float4* __restrict__ p, int n4) {
    int i = blockIdx.x * blockDim.x + threadIdx.x;
    if (i < n4) p[i] = make_float4(0.f, 0.f, 0.f, 0.f);
}

// ---------------------------------------------------------------------------
// Kernel 3: edge scatter-add. One wave32 per edge; lane l moves 4 floats.
// x row gather is a coalesced 512B read; agg update is 4 fp32 HW atomics
// (GLOBAL_ATOMIC_ADD_F32, non-returning -> STOREcnt, no CAS loop).
// ---------------------------------------------------------------------------
__device__ __forceinline__ void hw_atomic_add_f32(float* p, float v) {
    asm volatile("global_atomic_add_f32 %0, %1, off"
                 :
                 : "v"(p), "v"(v)
                 : "memory");
}

__global__ __launch_bounds__(256) void scatter_add_kernel(
        const float* __restrict__ x,
        const int*   __restrict__ ei,
        float*       __restrict__ agg) {
    const int lane   = threadIdx.x & 31;
    const int wave   = blockIdx.x * (blockDim.x >> 5) + (threadIdx.x >> 5);
    const int nwaves = gridDim.x * (blockDim.x >> 5);
    for (int e = wave; e < N_EDGES; e += nwaves) {
        const int src = ei[e];
        const int dst = ei[N_EDGES + e];
        const float4 v = *(const float4*)(x + src * DIM + lane * 4);
        float* p = agg + dst * DIM + lane * 4;
        hw_atomic_add_f32(p + 0, v.x);
        hw_atomic_add_f32(p + 1, v.y);
        hw_atomic_add_f32(p + 2, v.z);
        hw_atomic_add_f32(p + 3, v.w);
    }
}

// ---------------------------------------------------------------------------
// Kernel 4: fused dual GEMM with V_WMMA_F32_16X16X4_F32.
//   out[16 nodes x 128 cols per block] = agg@W1^T + x@W2^T + bc
// 8 waves/block; wave w owns output columns [16w, 16w+16).
// Two independent accumulator chains (one per GEMM) so adjacent WMMAs have
// no RAW dependence; summed once at the end.
// ---------------------------------------------------------------------------
__global__ __launch_bounds__(256) void sage_gemm_kernel(
        const float* __restrict__ agg,
        const float* __restrict__ x,
        const float* __restrict__ W1,
        const float* __restrict__ W2,
        const float* __restrict__ bc,
        float*       __restrict__ out,
        int ntiles) {
    const int lane  = threadIdx.x & 31;
    const int wave  = threadIdx.x >> 5;
    const int n0    = wave * 16;
    const int ncol  = n0 + (lane & 15);          // output column this lane touches
    const int khalf = (lane >> 4) * 2;           // 0 or 2

    // Preload both weight matrices' B-fragments into VGPRs (reused for all tiles).
    v2f b1[32], b2[32];
#pragma unroll
    for (int s = 0; s < 32; ++s) {
        const int k = 4 * s + khalf;
        b1[s] = *(const v2f*)(W1 + ncol * DIM + k);
        b2[s] = *(const v2f*)(W2 + ncol * DIM + k);
    }
    const float bias = bc[ncol];

    for (int t = blockIdx.x; t < ntiles; t += gridDim.x) {
        const int row = t * 16 + (lane & 15);
        const float* pa = agg + row * DIM + khalf;
        const float* px = x   + row * DIM + khalf;

        v8f c1 = {bias, bias, bias, bias, bias, bias, bias, bias};
        v8f c2 = {0.f, 0.f, 0.f, 0.f, 0.f, 0.f, 0.f, 0.f};
#pragma unroll
        for (int s = 0; s < 32; ++s) {
            v2f a1 = *(const v2f*)(pa + 4 * s);
            v2f a2 = *(const v2f*)(px + 4 * s);
            c1 = __builtin_amdgcn_wmma_f32_16x16x4_f32(
                     false, a1, false, b1[s], (short)0, c1, false, false);
            c2 = __builtin_amdgcn_wmma_f32_16x16x4_f32(
                     false, a2, false, b2[s], (short)0, c2, false, false);
        }
        const v8f c = c1 + c2;

        const int mbase = t * 16 + ((lane >> 4) << 3);
#pragma unroll
        for (int r = 0; r < 8; ++r) {
            out[(mbase + r) * DIM + ncol] = c[r];
        }
    }
}

// ---------------------------------------------------------------------------
// Host-side launcher
// ---------------------------------------------------------------------------
extern "C" void kernel_launch(void* const* d_in, const int* in_sizes, int n_in,
                              void* d_out, int out_size, void* d_ws, size_t ws_size,
                              hipStream_t stream) {
    const float* x    = (const float*)d_in[0];
    const int*   ei   = (const int*)  d_in[1];
    const float* Wl   = (const float*)d_in[2];
    const float* bl   = (const float*)d_in[3];
    const float* Wr   = (const float*)d_in[4];
    const float* Wres = (const float*)d_in[5];
    const float* bres = (const float*)d_in[6];
    float* out = (float*)d_out;

    // Workspace layout (floats): agg | W1 | W2 | bc
    float* ws  = (float*)d_ws;
    float* agg = ws;                              // N_NODES*DIM
    float* W1  = agg + (size_t)N_NODES * DIM;     // DIM*DIM
    float* W2  = W1 + DIM * DIM;                  // DIM*DIM
    float* bcp = W2 + DIM * DIM;                  // DIM

    // 1) fold blend into weights
    combine_weights_kernel<<<(DIM * DIM + 255) / 256, 256, 0, stream>>>(
        Wl, bl, Wr, Wres, bres, W1, W2, bcp);

    // 2) zero agg
    const int n4 = (N_NODES * DIM) / 4;
    zero_kernel<<<(n4 + 255) / 256, 256, 0, stream>>>((float4*)agg, n4);

    // 3) scatter-add over edges (one wave per edge, grid-stride)
    scatter_add_kernel<<<4096, 256, 0, stream>>>(x, ei, agg);

    // 4) fused dual WMMA GEMM + bias
    const int ntiles = N_NODES / 16;              // 6250, exact
    sage_gemm_kernel<<<1280, 256, 0, stream>>>(agg, x, W1, W2, bcp, out, ntiles);
}